// DeepSpeedInferenceAttention_84043920048286
// MI455X (gfx1250) — compile-verified
//
#include <hip/hip_runtime.h>

typedef float v2f __attribute__((ext_vector_type(2)));
typedef float v8f __attribute__((ext_vector_type(8)));

constexpr int B_      = 16;
constexpr int D_MODEL = 2048;
constexpr int N_HEADS = 16;
constexpr int MAX_SEQ = 4096;
constexpr int HD      = 128;
constexpr int NWAVES  = 8;
constexpr int BLOCK   = NWAVES * 32;
constexpr int MAX_LEN = MAX_SEQ + 16;   // scores scratch (total_len <= MAX_SEQ)

__global__ __launch_bounds__(BLOCK)
void attn_decode_f32wmma(const float* __restrict__ query,
                         const float* __restrict__ key,
                         const float* __restrict__ value,
                         const float* __restrict__ kcache,
                         const float* __restrict__ vcache,
                         const int*   __restrict__ cache_len_p,
                         float*       __restrict__ out)
{
    __shared__ float sc[MAX_LEN];   // scores -> exp weights
    __shared__ float lq[HD];        // query row for this (b,h)
    __shared__ float red[BLOCK];    // block reductions
    __shared__ float lo[HD];        // output accumulator

    const int bh   = blockIdx.x;
    const int b    = bh / N_HEADS;
    const int h    = bh % N_HEADS;
    const int tid  = threadIdx.x;
    const int lane = tid & 31;      // wave32
    const int wave = tid >> 5;
    const int n16  = lane & 15;
    const int half = lane >> 4;

    const int cache_len  = cache_len_p[0];
    const int total      = cache_len + 1;          // + the freshly appended K/V row
    const int ntiles     = cache_len >> 4;
    const int tail_start = ntiles << 4;

    const size_t bh_off = (size_t)bh * MAX_SEQ * HD;
    const float* kc   = kcache + bh_off;
    const float* vc   = vcache + bh_off;
    const int    qoff = b * D_MODEL + h * HD;
    const float* knew = key   + qoff;              // split_heads of new key/value
    const float* vnew = value + qoff;

    if (tid < HD) lq[tid] = query[qoff + tid];
    __syncthreads();

    const float scale = 0.08838834764831845f;      // 1/sqrt(128)

    // ---- Phase 1: scores[i] = scale * dot(q, K[i]) via V_WMMA_F32_16X16X4_F32.
    // A (16x4 f32): lane L holds A[L%16][ (L/16)*2 + {0,1} ] -> replicate q over M rows.
    // B (4x16 f32): lane n holds B[{0,1}][n], lane n+16 holds B[{2,3}][n] -> key rows.
    // D row 0, col n = score of key n; rows replicated, so acc[0] of lane n is the score.
    for (int t = wave; t < ntiles; t += NWAVES) {
        const int ks = t << 4;
        const float* kp = kc + (size_t)(ks + n16) * HD + half * 2;
        const float* qp = lq + half * 2;
        v8f acc = {};
#pragma unroll
        for (int c = 0; c < 32; ++c) {             // 32 chunks of K=4 cover 128 dims
            v2f a  = *(const v2f*)(qp + 4 * c);    // ds_load_b64 (broadcast q chunk)
            v2f bb = *(const v2f*)(kp + 4 * c);    // global_load_b64 (key chunk)
            acc = __builtin_amdgcn_wmma_f32_16x16x4_f32(
                      false, a, false, bb, (short)0, acc, false, false);
        }
        if (lane < 16) sc[ks + lane] = acc[0] * scale;
    }

    // Ragged tail + the new key at index cache_len (<= 16 rows), wave 0 scalar dots.
    if (wave == 0) {
        const int r = tail_start + lane;
        if (r <= cache_len) {
            const float* rowp = (r < cache_len) ? (kc + (size_t)r * HD) : knew;
            float d = 0.f;
#pragma unroll 4
            for (int i = 0; i < HD; ++i) d += lq[i] * rowp[i];
            sc[r] = d * scale;
        }
    }
    __syncthreads();

    // ---- Phase 2: softmax over sc[0..total)
    float lm = -3.402823466e38f;
    for (int i = tid; i < total; i += BLOCK) lm = fmaxf(lm, sc[i]);
    red[tid] = lm;
    __syncthreads();
    for (int s = BLOCK / 2; s > 0; s >>= 1) {
        if (tid < s) red[tid] = fmaxf(red[tid], red[tid + s]);
        __syncthreads();
    }
    const float m = red[0];
    __syncthreads();

    float lsum = 0.f;
    for (int i = tid; i < total; i += BLOCK) {
        const float e = __expf(sc[i] - m);
        sc[i] = e;
        lsum += e;
    }
    red[tid] = lsum;
    __syncthreads();
    for (int s = BLOCK / 2; s > 0; s >>= 1) {
        if (tid < s) red[tid] += red[tid + s];
        __syncthreads();
    }
    const float inv = 1.0f / red[0];

    if (tid < HD) lo[tid] = 0.f;
    __syncthreads();

    // ---- Phase 3: out[d] = sum_i w_i * V[i][d]; one V row per wave per step,
    // float4 per lane -> global_load_b128, fully coalesced 512B/row.
    float a0 = 0.f, a1 = 0.f, a2 = 0.f, a3 = 0.f;
    for (int r = wave; r < total; r += NWAVES) {
        const float  w  = sc[r];
        const float* vp = (r < cache_len) ? (vc + (size_t)r * HD) : vnew;
        const float4 v4 = *(const float4*)(vp + lane * 4);
        a0 += w * v4.x; a1 += w * v4.y; a2 += w * v4.z; a3 += w * v4.w;
    }
    atomicAdd(&lo[lane * 4 + 0], a0);              // ds_add_f32
    atomicAdd(&lo[lane * 4 + 1], a1);
    atomicAdd(&lo[lane * 4 + 2], a2);
    atomicAdd(&lo[lane * 4 + 3], a3);
    __syncthreads();

    if (tid < HD) out[qoff + tid] = lo[tid] * inv;
}

extern "C" void kernel_launch(void* const* d_in, const int* in_sizes, int n_in,
                              void* d_out, int out_size, void* d_ws, size_t ws_size,
                              hipStream_t stream)
{
    (void)in_sizes; (void)n_in; (void)out_size; (void)d_ws; (void)ws_size;
    const float* q  = (const float*)d_in[0];
    const float* k  = (const float*)d_in[1];
    const float* v  = (const float*)d_in[2];
    const float* kc = (const float*)d_in[3];
    const float* vc = (const float*)d_in[4];
    const int*   cl = (const int*)  d_in[5];
    attn_decode_f32wmma<<<dim3(B_ * N_HEADS), dim3(BLOCK), 0, stream>>>(
        q, k, v, kc, vc, cl, (float*)d_out);
}